// Quan2d_72739566125681
// MI455X (gfx1250) — compile-verified
//
#include <hip/hip_runtime.h>

// Quanv2d via WMMA on gfx1250 (MI455X).
//
// Math: out[b, p*4+q] = sum_i sign(q,i) * (U @ psi0(b,p))_i^2
//   psi0 = tensor product of per-qubit 2-vectors from patch angles
//   U    = fixed 16x16 orthogonal matrix from `weights` (precomputed on device)
//
// Roofline: 12.8 MB in + 12.8 MB out = 25.6 MB -> ~1.1 us @ 23.3 TB/s.
// Compute (~0.4 GFLOP f32 WMMA + 3.2M sincos) is far below that, so the
// kernel is memory-bound; f32 WMMA keeps exact reference precision for free.
// Prep kernel uses hardware v_sin/v_cos so its single serial wave stays far
// off the critical path (precise OCML sincos previously cost ~6k serial
// instructions).

typedef __attribute__((ext_vector_type(2))) float v2f;
typedef __attribute__((ext_vector_type(8))) float v8f;

#define PATCHES      196   // 14*14 patches per image
#define OUT_STRIDE   784   // 196*4 outputs per image
#define INV_SQRT2    0.70710678118654752440f

// ---------------------------------------------------------------------------
// Prep kernel: build U (16x16, row-major) from weights (2 x 8 floats).
// Lane j evolves basis column e_j through the weight circuit: U[:,j] = G e_j.
// Qubit q maps to bit (3-q) of the flat basis index (mask = 8 >> q).
// ---------------------------------------------------------------------------
__device__ __forceinline__ void apply_ry16(float v[16], int st, float c, float s) {
  #pragma unroll
  for (int i = 0; i < 16; ++i) {
    if (!(i & st)) {
      const int i1 = i | st;
      const float a0 = v[i], a1 = v[i1];
      v[i]  = c * a0 - s * a1;   // RY = [[c,-s],[s,c]]
      v[i1] = s * a0 + c * a1;
    }
  }
}

__device__ __forceinline__ void apply_cnot16(float v[16], int cb, int tb) {
  #pragma unroll
  for (int i = 0; i < 16; ++i) {
    if ((i & cb) && !(i & tb)) {
      const int i1 = i | tb;
      const float t = v[i]; v[i] = v[i1]; v[i1] = t;
    }
  }
}

__global__ void build_unitary_kernel(const float* __restrict__ w,
                                     float* __restrict__ U) {
  const int j = threadIdx.x;
  if (j >= 16) return;              // no WMMA here, partial EXEC is fine

  // Hardware-transcendental sincos of all 16 weight half-angles (args <= pi).
  float wc[16], ws[16];
  #pragma unroll
  for (int i = 0; i < 16; ++i) {
    __sincosf(w[i] * 0.5f, &ws[i], &wc[i]);
  }

  float v[16];
  #pragma unroll
  for (int i = 0; i < 16; ++i) v[i] = (i == j) ? 1.0f : 0.0f;

  #pragma unroll
  for (int d = 0; d < 2; ++d) {
    #pragma unroll
    for (int q = 0; q < 4; ++q)
      apply_ry16(v, 8 >> q, wc[d * 8 + q], ws[d * 8 + q]);
    apply_cnot16(v, 8 >> 0, 8 >> 1);   // CNOT(0,1)
    apply_cnot16(v, 8 >> 1, 8 >> 2);   // CNOT(1,2)
    apply_cnot16(v, 8 >> 2, 8 >> 3);   // CNOT(2,3)
    #pragma unroll
    for (int q = 0; q < 4; ++q)
      apply_ry16(v, 8 >> q, wc[d * 8 + 4 + q], ws[d * 8 + 4 + q]);
  }
  #pragma unroll
  for (int i = 0; i < 16; ++i) U[i * 16 + j] = v[i];   // row-major U
}

// ---------------------------------------------------------------------------
// Main kernel: one wave = 16 circuits. Y = U * Psi0 via 4x V_WMMA_F32_16X16X4.
//
// VGPR layouts (ISA 7.12.2, 32-bit operands, wave32):
//   A (16x4, MxK): lane l -> row m = l&15; VGPR0 = K{0|2}, VGPR1 = K{1|3},
//                  high half-lanes carry K=2,3.  (chunk t covers K = 4t..4t+3)
//   B (4x16, KxN): mirrored — lane l -> col n = l&15; VGPR0 = K{0|2}, etc.
//   C/D (16x16):   lane l -> col n = l&15; VGPR v -> row m = v + 8*(l>=16).
// ---------------------------------------------------------------------------
__global__ __launch_bounds__(256) void quanv_wmma_kernel(
    const float* __restrict__ x,    // (4096, 1, 28, 28)
    const float* __restrict__ U,    // 16x16 row-major
    float* __restrict__ out) {      // (4096, 784)
  const unsigned lane = threadIdx.x & 31u;
  const unsigned wave = (blockIdx.x * 256u + threadIdx.x) >> 5;
  const unsigned h    = lane >> 4;       // which half of the wave
  const unsigned n    = lane & 15u;      // circuit within 16-wide tile
  const unsigned g    = wave * 16u + n;  // global circuit id (exact, no tail)

  const unsigned b  = g / 196u;          // image
  const unsigned p  = g - b * 196u;      // patch
  const unsigned pi = p / 14u;
  const unsigned pj = p - pi * 14u;

  // ---- load 2x2 patch (two contiguous float2 rows) --------------------
  const float* px = x + (size_t)b * 784u + (pi * 2u) * 28u + pj * 2u;
  const float2 r0 = *(const float2*)(px);
  const float2 r1 = *(const float2*)(px + 28);
  const float ang[4] = { r0.x, r0.y, r1.x, r1.y };

  // ---- per-qubit 2-vectors: RY(theta) H |0> ---------------------------
  float u0[4], u1[4];
  #pragma unroll
  for (int q = 0; q < 4; ++q) {
    float s, c;
    __sincosf(ang[q] * 0.5f, &s, &c);
    u0[q] = (c - s) * INV_SQRT2;
    u1[q] = (c + s) * INV_SQRT2;
  }

  // ---- psi0 = u_q0 (x) u_q1 (x) u_q2 (x) u_q3  (flat i = b0*8+b1*4+b2*2+b3)
  float t01[4], t23[4], psi[16];
  t01[0] = u0[0] * u0[1]; t01[1] = u0[0] * u1[1];
  t01[2] = u1[0] * u0[1]; t01[3] = u1[0] * u1[1];
  t23[0] = u0[2] * u0[3]; t23[1] = u0[2] * u1[3];
  t23[2] = u1[2] * u0[3]; t23[3] = u1[2] * u1[3];
  #pragma unroll
  for (int i = 0; i < 16; ++i) psi[i] = t01[i >> 2] * t23[i & 3];

  // ---- pack A (U slices) and B (psi slices) per chunk t ---------------
  const unsigned m  = n;           // A row for this lane
  const unsigned k0 = 2u * h;      // K offset carried by this half
  v2f a[4], bb[4];
  #pragma unroll
  for (int t = 0; t < 4; ++t) {
    const unsigned kk = 4u * t + k0;
    a[t].x  = U[m * 16u + kk];
    a[t].y  = U[m * 16u + kk + 1u];
    bb[t].x = psi[kk];
    bb[t].y = psi[kk + 1u];
  }

  // ---- Y = U * Psi0 : 4 chained f32 WMMAs (K = 16 total) --------------
  v8f acc = {0.f, 0.f, 0.f, 0.f, 0.f, 0.f, 0.f, 0.f};
  #pragma unroll
  for (int t = 0; t < 4; ++t) {
    acc = __builtin_amdgcn_wmma_f32_16x16x4_f32(
        /*neg_a=*/false, a[t], /*neg_b=*/false, bb[t],
        /*c_mod=*/(short)0, acc, /*reuse_a=*/false, /*reuse_b=*/false);
  }

  // ---- ev_q = sum_i sign(q,i) * y_i^2 ; rows split across lane halves -
  float ev[4];
  #pragma unroll
  for (int q = 0; q < 4; ++q) {
    float sum = 0.f;
    #pragma unroll
    for (int v = 0; v < 8; ++v) {
      const unsigned mm = (unsigned)v + 8u * h;   // basis row this VGPR holds
      const float y  = acc[v];
      const float yy = y * y;
      sum += ((mm >> (3 - q)) & 1u) ? -yy : yy;
    }
    ev[q] = sum + __shfl_xor(sum, 16, 32);        // combine the two halves
  }

  if (h == 0) {
    float4 o = { ev[0], ev[1], ev[2], ev[3] };
    *(float4*)(out + (size_t)b * OUT_STRIDE + p * 4u) = o;   // 16B aligned
  }
}

// ---------------------------------------------------------------------------
extern "C" void kernel_launch(void* const* d_in, const int* in_sizes, int n_in,
                              void* d_out, int out_size, void* d_ws, size_t ws_size,
                              hipStream_t stream) {
  const float* x = (const float*)d_in[0];   // 4096*1*28*28 f32
  const float* w = (const float*)d_in[1];   // 2*8 f32
  float* U   = (float*)d_ws;                // 16*16 f32 scratch
  float* out = (float*)d_out;               // 4096*784 f32

  build_unitary_kernel<<<1, 32, 0, stream>>>(w, U);

  // 4096*196 = 802816 circuits = 50176 waves = 6272 blocks of 8 waves. Exact.
  quanv_wmma_kernel<<<6272, 256, 0, stream>>>(x, U, out);
}